// LearnPose_45844480918144
// MI455X (gfx1250) — compile-verified
//
#include <hip/hip_runtime.h>
#include <stdint.h>

typedef float v4f __attribute__((ext_vector_type(4)));

#define TILE 256

__global__ __launch_bounds__(TILE) void pose_c2w_kernel(
    const float* __restrict__ r,
    const float* __restrict__ t,
    const float* __restrict__ c2w_init,
    float* __restrict__ out,
    int n)
{
    __shared__ __attribute__((aligned(16))) float s_rt[TILE * 6];

    const int tid = threadIdx.x;
    const int tile_start = blockIdx.x * TILE;
    const int rem = n - tile_start;
    const int count = rem < TILE ? rem : TILE;

    float rx, ry, rz, tx, ty, tz;

    if (count == TILE) {
        // Bulk async stage of this tile's r[] and t[] (TILE*12 B each) into LDS:
        // 192 lanes x 16 B per array, contiguous, ASYNCcnt-tracked (gfx1250).
        const unsigned lds_base = (unsigned)(uintptr_t)(void*)&s_rt[0];
        if (tid < (TILE * 12) / 16) {
            const unsigned voff  = (unsigned)tile_start * 12u + (unsigned)tid * 16u;
            const unsigned lds_r = lds_base + (unsigned)tid * 16u;
            const unsigned lds_t = lds_base + (unsigned)(TILE * 12) + (unsigned)tid * 16u;
            asm volatile("global_load_async_to_lds_b128 %0, %1, %2"
                         :: "v"(lds_r), "v"(voff), "s"((uint64_t)(uintptr_t)r)
                         : "memory");
            asm volatile("global_load_async_to_lds_b128 %0, %1, %2"
                         :: "v"(lds_t), "v"(voff), "s"((uint64_t)(uintptr_t)t)
                         : "memory");
        }
        asm volatile("s_wait_asynccnt 0" ::: "memory");
        __syncthreads();
        // Readback via explicit ds_load so the data path is visible to the
        // backend as asm outputs (cannot be folded away). Stride-3 dword
        // addresses: 32 lanes * 3 banks apart -> conflict-free.
        const unsigned addr_r = lds_base + (unsigned)tid * 12u;
        const unsigned addr_t = addr_r + (unsigned)(TILE * 12);
        asm volatile("ds_load_b32 %0, %6\n\t"
                     "ds_load_b32 %1, %6 offset:4\n\t"
                     "ds_load_b32 %2, %6 offset:8\n\t"
                     "ds_load_b32 %3, %7\n\t"
                     "ds_load_b32 %4, %7 offset:4\n\t"
                     "ds_load_b32 %5, %7 offset:8\n\t"
                     "s_wait_dscnt 0"
                     : "=&v"(rx), "=&v"(ry), "=&v"(rz),
                       "=&v"(tx), "=&v"(ty), "=&v"(tz)
                     : "v"(addr_r), "v"(addr_t)
                     : "memory");
    } else {
        if (tid >= count) return;
        const int cam = tile_start + tid;
        rx = r[cam * 3 + 0]; ry = r[cam * 3 + 1]; rz = r[cam * 3 + 2];
        tx = t[cam * 3 + 0]; ty = t[cam * 3 + 1]; tz = t[cam * 3 + 2];
    }

    const int cam = tile_start + tid;

    // Closed form of M = fixup(Rx @ Ry @ Rz):
    //   R row0 = [cy*cz, -cy*sz, sy]
    //   R row1 = [cx*sz + sx*sy*cz, cx*cz - sx*sy*sz, -sx*cy]
    //   R row2 = [sx*sz - cx*sy*cz, sx*cz + cx*sy*sz,  cx*cy]
    //   M = [[R00,-R10,R20], [-R01,R11,-R21], [-R02,-R12,R22]]
    float sx, cx, sy, cy, sz, cz;
    __sincosf(rx, &sx, &cx);
    __sincosf(ry, &sy, &cy);
    __sincosf(rz, &sz, &cz);

    const float sxsy = sx * sy;
    const float cxsy = cx * sy;

    v4f row0 = { cy * cz,
                 -(cx * sz + sxsy * cz),
                 sx * sz - cxsy * cz,
                 tx * 3840.0f };
    v4f row1 = { cy * sz,
                 cx * cz - sxsy * sz,
                 -(sx * cz + cxsy * sz),
                 ty * 2160.0f };
    v4f row2 = { -sy,
                 sx * cy,
                 cx * cy,
                 tz };
    // Bottom row passes through from c2w_init; single-use -> non-temporal.
    v4f row3 = __builtin_nontemporal_load(
        (const v4f*)(c2w_init + (size_t)cam * 16 + 12));

    float* o = out + (size_t)cam * 16;
    __builtin_nontemporal_store(row0, (v4f*)(o + 0));
    __builtin_nontemporal_store(row1, (v4f*)(o + 4));
    __builtin_nontemporal_store(row2, (v4f*)(o + 8));
    __builtin_nontemporal_store(row3, (v4f*)(o + 12));
}

extern "C" void kernel_launch(void* const* d_in, const int* in_sizes, int n_in,
                              void* d_out, int out_size, void* d_ws, size_t ws_size,
                              hipStream_t stream) {
    const float* r        = (const float*)d_in[0];
    const float* t        = (const float*)d_in[1];
    const float* c2w_init = (const float*)d_in[2];
    float* out = (float*)d_out;
    const int n = in_sizes[0] / 3;
    const int blocks = (n + TILE - 1) / TILE;
    hipLaunchKernelGGL(pose_c2w_kernel, dim3(blocks), dim3(TILE), 0, stream,
                       r, t, c2w_init, out, n);
}